// GSCMessagePassing_50800873177305
// MI455X (gfx1250) — compile-verified
//
#include <hip/hip_runtime.h>
#include <hip/hip_bf16.h>
#include <math.h>

#define NUM_EDGE_TYPES 38
#define NUM_NODE_TYPES 4
#define HIDDEN 64
#define COMBOS (NUM_EDGE_TYPES * NUM_NODE_TYPES * NUM_NODE_TYPES) // 608

typedef float v2f __attribute__((ext_vector_type(2)));
typedef float v8f __attribute__((ext_vector_type(8)));

// ---------------------------------------------------------------------------
// Kernel 1: build the 608-entry edge-embedding table with f32 WMMA.
// One wave (32 threads) per 16-combo tile; 38 blocks total.
// h[row][0..63] staged in LDS, then D = h @ W2 via 16 x V_WMMA_F32_16X16X4_F32.
// ---------------------------------------------------------------------------
__global__ __launch_bounds__(32) void gsc_table_kernel(
    const float* __restrict__ W1, const float* __restrict__ b1,
    const float* __restrict__ W2, const float* __restrict__ b2,
    float* __restrict__ table) {
  __shared__ float hs[16][68];  // pad 64->68: stride%64==4 -> conflict-free col reads

  const int tile = blockIdx.x;       // edge_type
  const int lane = threadIdx.x;      // 0..31, one full wave
  const int row  = lane & 15;        // combo-in-tile = head*4 + tail
  const int half = lane >> 4;        // 0 or 1
  const int ht = row >> 2, tt = row & 3;

  const float* r0 = W1 + tile * HIDDEN;
  const float* r1 = W1 + (NUM_EDGE_TYPES + ht) * HIDDEN;
  const float* r2 = W1 + (NUM_EDGE_TYPES + NUM_NODE_TYPES + tt) * HIDDEN;

  // Each lane computes 32 of the 64 hidden values for its row (tanh-GELU).
  const int jbase = half * 32;
#pragma unroll 4
  for (int j = jbase; j < jbase + 32; ++j) {
    float x  = r0[j] + r1[j] + r2[j] + b1[j];
    float t  = tanhf(0.7978845608028654f * (x + 0.044715f * x * x * x));
    hs[row][j] = 0.5f * x * (1.0f + t);
  }
  __syncthreads();

  // Accumulate D = A(16x4) * B(4x16) over 16 K-chunks, exact f32.
  // A layout: lanes 0-15 -> K = {0,1} in V0/V1; lanes 16-31 -> K = {2,3}.
  // B layout mirrors A in K; broadcast W2[k] along N so column 0 = dot product.
  v8f c = {};
  const int koff = half * 2;
#pragma unroll
  for (int kk = 0; kk < HIDDEN; kk += 4) {
    v2f a, b;
    a.x = hs[row][kk + koff];
    a.y = hs[row][kk + koff + 1];
    b.x = W2[kk + koff];
    b.y = W2[kk + koff + 1];
    c = __builtin_amdgcn_wmma_f32_16x16x4_f32(
        /*neg_a=*/false, a, /*neg_b=*/false, b,
        /*c_mod=*/(short)0, c, /*reuse_a=*/false, /*reuse_b=*/false);
  }

  // C/D layout: lane L (N = L%16), VGPR v -> M = v + 8*(L/16).
  // Lanes 0 and 16 hold column N=0, i.e. the dot products.
  if ((lane & 15) == 0) {
    const int mbase = half * 8;
    const float bias = b2[0];
#pragma unroll
    for (int v = 0; v < 8; ++v) {
      float dot = c[v] + bias;
      table[tile * 16 + mbase + v] = 1.0f / (1.0f + expf(-dot));
    }
  }
}

// ---------------------------------------------------------------------------
// Kernel 2: per-edge embedding = table lookup (replaces the 1.6M-edge MLP).
// ---------------------------------------------------------------------------
__global__ __launch_bounds__(256) void gsc_edge_kernel(
    const int* __restrict__ src, const int* __restrict__ dst,
    const int* __restrict__ etype, const int* __restrict__ ntype,
    const float* __restrict__ table, float* __restrict__ emb, int E) {
  int e = blockIdx.x * blockDim.x + threadIdx.x;
  if (e >= E) return;
  int combo = etype[e] * 16 + ntype[src[e]] * 4 + ntype[dst[e]];
  emb[e] = table[combo];
}

__global__ __launch_bounds__(256) void gsc_zero_kernel(float* __restrict__ p, int n) {
  int i = blockIdx.x * blockDim.x + threadIdx.x;
  if (i < n) p[i] = 0.0f;
}

// Hop 1: prev aggr is all-zero, so msg == emb.
__global__ __launch_bounds__(256) void gsc_hop_first_kernel(
    const int* __restrict__ dst, const float* __restrict__ emb,
    float* __restrict__ out, int E) {
  int e = blockIdx.x * blockDim.x + threadIdx.x;
  if (e >= E) return;
  atomicAdd(&out[dst[e]], emb[e]);
}

// Hops 2..4: msg = prev[src] + emb, scatter-add at dst.
__global__ __launch_bounds__(256) void gsc_hop_kernel(
    const int* __restrict__ src, const int* __restrict__ dst,
    const float* __restrict__ emb, const float* __restrict__ prev,
    float* __restrict__ next, int E) {
  int e = blockIdx.x * blockDim.x + threadIdx.x;
  if (e >= E) return;
  atomicAdd(&next[dst[e]], prev[src[e]] + emb[e]);
}

extern "C" void kernel_launch(void* const* d_in, const int* in_sizes, int n_in,
                              void* d_out, int out_size, void* d_ws, size_t ws_size,
                              hipStream_t stream) {
  const int* edge_index = (const int*)d_in[0];   // (2, E)
  const int* edge_type  = (const int*)d_in[1];   // (E,)
  const int* node_type  = (const int*)d_in[2];   // (N,)
  const float* W1 = (const float*)d_in[3];       // (46, 64)
  const float* b1 = (const float*)d_in[4];       // (64,)
  const float* W2 = (const float*)d_in[5];       // (64, 1)
  const float* b2 = (const float*)d_in[6];       // (1,)
  float* out = (float*)d_out;                    // (N, 1)

  const int E = in_sizes[1];
  const int N = in_sizes[2];
  const int* src = edge_index;
  const int* dst = edge_index + E;

  // Workspace layout (floats): [table 608 pad->1024 | emb E | bufA N | bufB N]
  float* table = (float*)d_ws;
  float* emb   = table + 1024;
  float* bufA  = emb + E;
  float* bufB  = bufA + N;

  const int TB = 256;
  const int egrid = (E + TB - 1) / TB;
  const int ngrid = (N + TB - 1) / TB;

  // 1) 608-combo table via f32 WMMA (38 waves).
  gsc_table_kernel<<<COMBOS / 16, 32, 0, stream>>>(W1, b1, W2, b2, table);

  // 2) Per-edge embedding lookup.
  gsc_edge_kernel<<<egrid, TB, 0, stream>>>(src, dst, edge_type, node_type,
                                            table, emb, E);

  // 3) Four hops of gather + scatter-add (ping-pong, final hop -> d_out).
  gsc_zero_kernel<<<ngrid, TB, 0, stream>>>(bufA, N);
  gsc_hop_first_kernel<<<egrid, TB, 0, stream>>>(dst, emb, bufA, E);

  gsc_zero_kernel<<<ngrid, TB, 0, stream>>>(bufB, N);
  gsc_hop_kernel<<<egrid, TB, 0, stream>>>(src, dst, emb, bufA, bufB, E);

  gsc_zero_kernel<<<ngrid, TB, 0, stream>>>(bufA, N);
  gsc_hop_kernel<<<egrid, TB, 0, stream>>>(src, dst, emb, bufB, bufA, E);

  gsc_zero_kernel<<<ngrid, TB, 0, stream>>>(out, N);
  gsc_hop_kernel<<<egrid, TB, 0, stream>>>(src, dst, emb, bufA, out, E);
}